// RNN_9972914061534
// MI455X (gfx1250) — compile-verified
//
#include <hip/hip_runtime.h>

typedef float v2f __attribute__((ext_vector_type(2)));
typedef float v8f __attribute__((ext_vector_type(8)));

namespace {

constexpr int kB  = 2048;
constexpr int kT  = 512;
constexpr int kIn = 28;    // IN_DIM
constexpr int kH  = 64;    // hidden
constexpr int kC  = 10;    // classes
constexpr int kBM = 16;    // batch rows per workgroup (one WMMA M-tile)

__device__ __forceinline__ v8f wmma4(v2f a, v2f b, v8f c) {
  // D = A(16x4 f32) * B(4x16 f32) + C(16x16 f32)
  return __builtin_amdgcn_wmma_f32_16x16x4_f32(
      /*neg_a=*/false, a, /*neg_b=*/false, b,
      /*c_mod=*/(short)0, c, /*reuse_a=*/false, /*reuse_b=*/false);
}

__device__ __forceinline__ float sigmoid_f(float x) {
  return 1.0f / (1.0f + __expf(-x));
}
__device__ __forceinline__ float tanh_f(float x) {
  // tanh(x) = 1 - 2/(exp(2x)+1); saturates correctly at +/-1
  return 1.0f - 2.0f / (__expf(2.0f * x) + 1.0f);
}

__global__ __launch_bounds__(128, 1) void gru_fused_kernel(
    const float* __restrict__ x,      // (B, T, 28)
    const float* __restrict__ W_ih,   // (192, 28)
    const float* __restrict__ W_hh,   // (192, 64)
    const float* __restrict__ b_ih,   // (192)
    const float* __restrict__ b_hh,   // (192)
    const float* __restrict__ W_out,  // (10, 64)
    const float* __restrict__ b_out,  // (10)
    float* __restrict__ out)          // (B, 10)
{
  __shared__ float h_lds[kBM * kH];   // 4 KB: h state exchanged between waves

  const int tid  = threadIdx.x;
  const int wave = tid >> 5;          // 0..3, owns h columns [16w, 16w+16)
  const int lane = tid & 31;
  const int m    = lane & 15;         // tile row (A) or tile col (B/C)
  const int hi   = lane >> 4;         // half-select within K groups of 4
  const int bm   = blockIdx.x * kBM;  // batch row base

  // ---- Pin all weight B-fragments in VGPRs (wave32: up to 1024 VGPRs) ----
  // B fragment for K-step s: lane holds W^T[4s+2hi][n], W^T[4s+2hi+1][n]
  //                        = consecutive pair of W[n] row -> single b64 load.
  v2f whhB[3][16];  // 3 gates x (K=64 / 4) steps
  v2f wihB[3][7];   // 3 gates x (K=28 / 4) steps
#pragma unroll
  for (int g = 0; g < 3; ++g) {
    const int nrow = g * 64 + wave * 16 + m;   // output column in 192-wide G
#pragma unroll
    for (int s = 0; s < 16; ++s)
      whhB[g][s] = *(const v2f*)(W_hh + nrow * kH + 4 * s + 2 * hi);
#pragma unroll
    for (int s = 0; s < 7; ++s)
      wihB[g][s] = *(const v2f*)(W_ih + nrow * kIn + 4 * s + 2 * hi);
  }

  // Biases broadcast per C-tile column (N = lane%16 within this wave's slice)
  const int col = wave * 16 + m;               // h column owned by this lane
  const float bias_r  = b_ih[col]       + b_hh[col];
  const float bias_z  = b_ih[64 + col]  + b_hh[64 + col];
  const float bias_xn = b_ih[128 + col];       // n-gate input half (with b_ih)
  const float bias_hn = b_hh[128 + col];       // n-gate hidden half (with b_hh)

  // h state: A-fragments (full 16x64 per wave) + C-layout copy of own columns
  v2f hA[16];
#pragma unroll
  for (int s = 0; s < 16; ++s) hA[s] = v2f{0.0f, 0.0f};
  v8f h_c = {};

  const float* xbase = x + (size_t)(bm + m) * (kT * kIn) + 2 * hi;

  for (int t = 0; t < kT; ++t) {
    // x_t A-fragments: 7 x b64 loads (K = 0..27 exactly)
    v2f xA[7];
    const float* xt = xbase + t * kIn;
#pragma unroll
    for (int s = 0; s < 7; ++s) xA[s] = *(const v2f*)(xt + 4 * s);

    v8f acc_r, acc_z, acc_xn, acc_hn;
#pragma unroll
    for (int j = 0; j < 8; ++j) {
      acc_r[j]  = bias_r;
      acc_z[j]  = bias_z;
      acc_xn[j] = bias_xn;
      acc_hn[j] = bias_hn;
    }

    // xg contribution: x_t(16x28) @ W_ih^T -> r, z, xn tiles
#pragma unroll
    for (int s = 0; s < 7; ++s) {
      acc_r  = wmma4(xA[s], wihB[0][s], acc_r);
      acc_z  = wmma4(xA[s], wihB[1][s], acc_z);
      acc_xn = wmma4(xA[s], wihB[2][s], acc_xn);
    }
    // gh contribution: h(16x64) @ W_hh^T -> r, z, hn tiles
#pragma unroll
    for (int s = 0; s < 16; ++s) {
      acc_r  = wmma4(hA[s], whhB[0][s], acc_r);
      acc_z  = wmma4(hA[s], whhB[1][s], acc_z);
      acc_hn = wmma4(hA[s], whhB[2][s], acc_hn);
    }

    // GRU elementwise update on this wave's 16x16 C tile
    v8f h_new;
#pragma unroll
    for (int j = 0; j < 8; ++j) {
      const float r = sigmoid_f(acc_r[j]);
      const float z = sigmoid_f(acc_z[j]);
      const float n = tanh_f(acc_xn[j] + r * acc_hn[j]);
      h_new[j] = (1.0f - z) * n + z * h_c[j];
    }
    h_c = h_new;

    // Exchange h through LDS (C layout -> A fragments), split barriers:
    __syncthreads();   // all waves finished reading previous h from LDS
#pragma unroll
    for (int j = 0; j < 8; ++j)
      h_lds[(j + 8 * hi) * kH + col] = h_new[j];   // row j+8*hi, own column
    __syncthreads();   // h_{t+1} fully visible
#pragma unroll
    for (int s = 0; s < 16; ++s)
      hA[s] = *(const v2f*)(h_lds + m * kH + 4 * s + 2 * hi);
  }

  // ---- Classifier + softmax: one thread per batch row (h_T is in LDS) ----
  if (tid < kBM) {
    float logits[kC];
    float mx = -3.4e38f;
#pragma unroll
    for (int c = 0; c < kC; ++c) {
      float d = b_out[c];
#pragma unroll
      for (int k = 0; k < kH; ++k)
        d += h_lds[tid * kH + k] * W_out[c * kH + k];
      logits[c] = d;
      mx = fmaxf(mx, d);
    }
    float sum = 0.0f;
#pragma unroll
    for (int c = 0; c < kC; ++c) {
      const float e = __expf(logits[c] - mx);
      logits[c] = e;
      sum += e;
    }
    const float inv = 1.0f / sum;
#pragma unroll
    for (int c = 0; c < kC; ++c)
      out[(size_t)(bm + tid) * kC + c] = logits[c] * inv;
  }
}

}  // namespace

extern "C" void kernel_launch(void* const* d_in, const int* in_sizes, int n_in,
                              void* d_out, int out_size, void* d_ws, size_t ws_size,
                              hipStream_t stream) {
  (void)in_sizes; (void)n_in; (void)out_size; (void)d_ws; (void)ws_size;
  const float* x     = (const float*)d_in[0];
  const float* W_ih  = (const float*)d_in[1];
  const float* W_hh  = (const float*)d_in[2];
  const float* b_ih  = (const float*)d_in[3];
  const float* b_hh  = (const float*)d_in[4];
  const float* W_out = (const float*)d_in[5];
  const float* b_out = (const float*)d_in[6];
  float* out = (float*)d_out;

  dim3 grid(kB / kBM);   // 128 workgroups, one 16-row batch tile each
  dim3 block(128);       // 4 wave32s: one per 16-column slice of h
  gru_fused_kernel<<<grid, block, 0, stream>>>(x, W_ih, W_hh, b_ih, b_hh,
                                               W_out, b_out, out);
}